// QuanvolutionHybrid_65481071404148
// MI455X (gfx1250) — compile-verified
//
#include <hip/hip_runtime.h>

#define NQ 4

typedef float v2f __attribute__((ext_vector_type(2)));
typedef float v8f __attribute__((ext_vector_type(8)));

// ---------------------------------------------------------------------------
// Kernel 1: build the 16x16 complex unitary U(theta) for the shared ansatz
// (3 x [RY layer, RZ layer] with 2 full-entanglement CNOT layers between).
// One block, 256 threads: thread t owns element (k = t>>4, j = t&15).
// Qubit q maps to bit (3-q) of the basis index (psi reshaped (2,2,2,2)).
// Output: d_ws[0..255] = U_re[k*16+j], d_ws[256..511] = U_im[k*16+j].
// ---------------------------------------------------------------------------
__global__ __launch_bounds__(256) void quanv_build_unitary(
    const float* __restrict__ theta, float* __restrict__ Uout) {
  __shared__ float Ure[16][16];
  __shared__ float Uim[16][16];
  const int t = threadIdx.x;
  const int k = t >> 4;
  const int j = t & 15;
  Ure[k][j] = (k == j) ? 1.0f : 0.0f;
  Uim[k][j] = 0.0f;
  __syncthreads();

  for (int r = 0; r < 3; ++r) {
    // RY(theta[r,0,q]) on each qubit: row0' = c*row0 - s*row1; row1' = s*row0 + c*row1
    for (int q = 0; q < NQ; ++q) {
      const float th = theta[(r * 2 + 0) * 4 + q];
      float s, c;
      __sincosf(th * 0.5f, &s, &c);
      const int m = 1 << (3 - q);
      const int kp = k ^ m;
      const float a_re = Ure[k][j], a_im = Uim[k][j];
      const float b_re = Ure[kp][j], b_im = Uim[kp][j];
      __syncthreads();
      if (k & m) {
        Ure[k][j] = s * b_re + c * a_re;
        Uim[k][j] = s * b_im + c * a_im;
      } else {
        Ure[k][j] = c * a_re - s * b_re;
        Uim[k][j] = c * a_im - s * b_im;
      }
      __syncthreads();
    }
    // RZ(theta[r,1,q]): multiply row by exp(-+ i*th/2) (bit=0 -> -th/2)
    for (int q = 0; q < NQ; ++q) {
      const float th = theta[(r * 2 + 1) * 4 + q];
      const int m = 1 << (3 - q);
      const float half = (k & m) ? 0.5f * th : -0.5f * th;
      float sp, cp;
      __sincosf(half, &sp, &cp);
      const float re = Ure[k][j], im = Uim[k][j];
      Ure[k][j] = re * cp - im * sp;
      Uim[k][j] = re * sp + im * cp;
      __syncthreads();
    }
    if (r < 2) {
      // Full entanglement: CX(c,t) for all pairs c<t. U' row k = U row CNOT(k).
      const int pc[6] = {0, 0, 0, 1, 1, 2};
      const int pt[6] = {1, 2, 3, 2, 3, 3};
      for (int e = 0; e < 6; ++e) {
        const int mc = 1 << (3 - pc[e]);
        const int mt = 1 << (3 - pt[e]);
        const int src = (k & mc) ? (k ^ mt) : k;
        const float re = Ure[src][j], im = Uim[src][j];
        __syncthreads();
        Ure[k][j] = re;
        Uim[k][j] = im;
        __syncthreads();
      }
    }
  }
  Uout[k * 16 + j] = Ure[k][j];
  Uout[256 + k * 16 + j] = Uim[k][j];
}

// ---------------------------------------------------------------------------
// Kernel 2: one wave32 <-> 16 consecutive 2x2 patches (32 pixel columns, 2 rows).
// Encoding product state psi (16 reals/patch) built from 4 sincos; then
// amp = U * Psi as two f32 WMMA GEMMs (re & im), 4 accumulating
// V_WMMA_F32_16X16X4_F32 each over the K=16 basis dimension.
// <Z_q> = sign-weighted reduction of |amp|^2 over D rows (+ shfl_xor(16)).
// ---------------------------------------------------------------------------
__global__ __launch_bounds__(256) void quanv_wmma_kernel(
    const float* __restrict__ x, const float* __restrict__ U,
    float* __restrict__ out) {
  const int lane = threadIdx.x & 31;
  const int wave = (blockIdx.x << 3) | (threadIdx.x >> 5);  // 16384 waves total
  const int jt = wave & 7;          // tile of 16 patches within patch-row
  const int i = (wave >> 3) & 127;  // patch row
  const int b = wave >> 10;         // image

  // Coalesced pixel loads: 32 consecutive columns from 2 image rows.
  const float* img = x + (size_t)b * 65536;
  const int col = (jt << 5) | lane;
  const float r0 = img[(2 * i) * 256 + col];
  const float r1 = img[(2 * i + 1) * 256 + col];

  const int p = lane & 15;  // patch owned by this lane (both wave halves)
  const bool hi = lane >= 16;

  // Gather the 4 angles of patch p via wave shuffles (pixels 2p, 2p+1).
  const float a0 = __shfl(r0, 2 * p, 32);
  const float a1 = __shfl(r0, 2 * p + 1, 32);
  const float a2 = __shfl(r1, 2 * p, 32);
  const float a3 = __shfl(r1, 2 * p + 1, 32);

  float s0, c0, s1, c1, s2, c2, s3, c3;
  __sincosf(0.5f * a0, &s0, &c0);
  __sincosf(0.5f * a1, &s1, &c1);
  __sincosf(0.5f * a2, &s2, &c2);
  __sincosf(0.5f * a3, &s3, &c3);

  // psi_j = t01[j>>2] * (q2 factor) * (q3 factor); bit(3-q) of j selects c/s.
  const float t01[4] = {c0 * c1, c0 * s1, s0 * c1, s0 * s1};
  const float f2 = hi ? s2 : c2;  // low lanes: j%4 in {0,1} -> q2 bit 0
  const float fx = f2 * c3;       // j%4 even -> q3 bit 0
  const float fy = f2 * s3;       // j%4 odd  -> q3 bit 1

  // A-matrix (16x4 f32) per K-chunk s: lane holds U[p][4s+ofs], U[p][4s+ofs+1]
  // (ofs = 0 for lanes 0-15, 2 for lanes 16-31), as float2 from the L2-hot table.
  const float2* Ure2 = (const float2*)U;
  const float2* Uim2 = (const float2*)(U + 256);
  const int rowbase = (p << 3) + (hi ? 1 : 0);

  v8f dre = {0.f, 0.f, 0.f, 0.f, 0.f, 0.f, 0.f, 0.f};
  v8f dim_ = {0.f, 0.f, 0.f, 0.f, 0.f, 0.f, 0.f, 0.f};
#pragma unroll
  for (int s = 0; s < 4; ++s) {
    const float2 ar = Ure2[rowbase + 2 * s];
    const float2 ai = Uim2[rowbase + 2 * s];
    const v2f a_re = {ar.x, ar.y};
    const v2f a_im = {ai.x, ai.y};
    const v2f bb = {t01[s] * fx, t01[s] * fy};
    dre = __builtin_amdgcn_wmma_f32_16x16x4_f32(false, a_re, false, bb,
                                                (short)0, dre, false, false);
    dim_ = __builtin_amdgcn_wmma_f32_16x16x4_f32(false, a_im, false, bb,
                                                 (short)0, dim_, false, false);
  }

  // D layout: VGPR r holds basis row k = r + (hi ? 8 : 0) for patch p.
  // <Z_q> = sum_k sign_q(k) * (re^2 + im^2), sign from bit (3-q) of k.
  const int base = hi ? 8 : 0;
  float z0 = 0.f, z1 = 0.f, z2 = 0.f, z3 = 0.f;
#pragma unroll
  for (int rr = 0; rr < 8; ++rr) {
    const float pr = dre[rr] * dre[rr] + dim_[rr] * dim_[rr];
    const int kk = base + rr;
    z0 += ((kk >> 3) & 1) ? -pr : pr;
    z1 += ((kk >> 2) & 1) ? -pr : pr;
    z2 += ((kk >> 1) & 1) ? -pr : pr;
    z3 += (kk & 1) ? -pr : pr;
  }
  // Combine the two wave halves (same patch, complementary basis rows).
  z0 += __shfl_xor(z0, 16, 32);
  z1 += __shfl_xor(z1, 16, 32);
  z2 += __shfl_xor(z2, 16, 32);
  z3 += __shfl_xor(z3, 16, 32);

  if (!hi) {
    const int patch = ((b << 7) + i) * 128 + (jt << 4) + p;  // b*16384 + i*128 + ...
    float4 o = make_float4(z0, z1, z2, z3);
    *(float4*)(out + (size_t)patch * 4) = o;  // 16B/patch, coalesced B128 stores
  }
}

// ---------------------------------------------------------------------------
extern "C" void kernel_launch(void* const* d_in, const int* in_sizes, int n_in,
                              void* d_out, int out_size, void* d_ws,
                              size_t ws_size, hipStream_t stream) {
  (void)in_sizes; (void)n_in; (void)out_size; (void)ws_size;
  const float* x = (const float*)d_in[0];      // (16,1,256,256) f32
  const float* theta = (const float*)d_in[1];  // (3,2,4) f32
  float* out = (float*)d_out;                  // (16, 65536) f32
  float* U = (float*)d_ws;                     // 512 floats: U_re | U_im

  quanv_build_unitary<<<1, 256, 0, stream>>>(theta, U);
  // 16 images * 128 patch-rows * 8 tiles = 16384 waves; 8 waves/block.
  quanv_wmma_kernel<<<2048, 256, 0, stream>>>(x, U, out);
}